// RegionLoss_65755949301936
// MI455X (gfx1250) — compile-verified
//
#include <hip/hip_runtime.h>
#include <math.h>

// ---- problem constants (match reference) ----
#define KC     9
#define NCLS   13
#define NAN_   5
#define NB_    128
#define NH_    26
#define NW_    26
#define NT_    50
#define LT_    21          // 2K+3
#define NPIX   (NH_*NW_)   // 676
#define NANCH  (NAN_*NPIX) // 3380
#define NCELLS (NB_*NANCH) // 432640
#define CHAN   (2*KC+1+NCLS) // 32

#define TH_D     80.0f
#define SHARP_   2.0f
#define SIL_     0.6f
#define NOOBJ_   1.0f
#define OBJ_     5.0f
#define SX2      409600.0f   // 640^2
#define SY2      230400.0f   // 480^2
#define INV_CONF0 0.15651764274966565f  // 1/(e^2-1)
#define PRETRAIN 15

#define LOG2E_   1.4426950408889634f
#define LN2_     0.6931471805599453f
// exp(SHARP*(1 - d/TH)) = exp2(C1 - C2*d)
#define C1_      (SHARP_ * LOG2E_)                 // 2.8853900817779268
#define C2_      ((SHARP_ / TH_D) * LOG2E_)        // 0.036067376022224085

#define CELL_TILES ((NANCH + 15) / 16)  // 212

typedef float v2f __attribute__((ext_vector_type(2)));
typedef float v8f __attribute__((ext_vector_type(8)));

// fast sigmoid: v_rcp + v_exp (adequate for a scalar loss)
__device__ __forceinline__ float sigmoidf_(float x) {
    return __builtin_amdgcn_rcpf(1.0f + __builtin_amdgcn_exp2f(-x * LOG2E_));
}

// corner confidence from squared scaled distance.
// |d2| instead of max(d2,0): negative d2 only arises from rounding
// cancellation (true d ~ 0), and fabs folds into v_sqrt's source modifier.
__device__ __forceinline__ float corner_c(float d2) {
    float d = __builtin_amdgcn_sqrtf(__builtin_fabsf(d2));
    float e = __builtin_amdgcn_exp2f(fmaf(-C2_, d, C1_));  // exp(SHARP*(1-d/TH))
    float c = fmaf(e, INV_CONF0, -INV_CONF0);              // (e-1)/CONF0
    return (d < TH_D) ? c : 0.0f;
}

// ---------------------------------------------------------------------------
// 0) zero-init scratch arrays that the scatter kernel updates sparsely + out
// ---------------------------------------------------------------------------
__global__ void init_kernel(float* coord_mask, float* cls_mask, float* tconf,
                            float* tcls, float* tx, float* ty, float* out) {
    long idx = (long)blockIdx.x * blockDim.x + threadIdx.x;
    if (idx < (long)NCELLS) {
        coord_mask[idx] = 0.0f; cls_mask[idx] = 0.0f;
        tconf[idx] = 0.0f;      tcls[idx] = 0.0f;
    }
    if (idx < (long)NCELLS * KC) { tx[idx] = 0.0f; ty[idx] = 0.0f; }
    if (idx == 0) out[0] = 0.0f;
}

// ---------------------------------------------------------------------------
// 1) predicted corner positions px,py  (layout [b][k][cell] for coalescing)
// ---------------------------------------------------------------------------
__global__ void compute_pred_kernel(const float* __restrict__ output,
                                    float* __restrict__ px, float* __restrict__ py) {
    long idx = (long)blockIdx.x * blockDim.x + threadIdx.x;
    if (idx >= (long)NCELLS) return;
    int b = (int)(idx / NANCH);
    int cell = (int)(idx % NANCH);
    int a = cell / NPIX;
    int pix = cell % NPIX;
    int j = pix / NW_, i = pix % NW_;
    const float* ob = output + ((long)(b * NAN_ + a) * CHAN) * NPIX + pix;
#pragma unroll
    for (int k = 0; k < KC; ++k) {
        float xs = (k == 0) ? sigmoidf_(ob[0]) : ob[(2 * k) * NPIX];
        float ys = (k == 0) ? sigmoidf_(ob[NPIX]) : ob[(2 * k + 1) * NPIX];
        px[((long)(b * KC + k)) * NANCH + cell] = (xs + (float)i) * (1.0f / (float)NW_);
        py[((long)(b * KC + k)) * NANCH + cell] = (ys + (float)j) * (1.0f / (float)NH_);
    }
}

// ---------------------------------------------------------------------------
// 2) per-target prep: valid count, gt coords, anchor argmax, conf_t, tx/ty
// ---------------------------------------------------------------------------
__global__ void prep_targets_kernel(const float* __restrict__ tgt,
                                    const float* __restrict__ px, const float* __restrict__ py,
                                    float* __restrict__ gtx, float* __restrict__ gty,
                                    int* __restrict__ nv,
                                    int* __restrict__ gi0a, int* __restrict__ gj0a,
                                    int* __restrict__ bestn,
                                    float* __restrict__ conf_t,
                                    float* __restrict__ txv, float* __restrict__ tyv,
                                    float* __restrict__ tclsv) {
    const float AW[NAN_] = {1.482f, 2.0501f, 2.3946f, 3.1018f, 3.4879f};
    const float AH[NAN_] = {2.2412f, 3.1265f, 4.6891f, 3.991f, 5.8851f};
    int b = blockIdx.x;
    const float* tb = tgt + (long)b * NT_ * LT_;
    if (threadIdx.x == 0) {
        int n = 0;
        for (int t = 0; t < NT_; ++t) {
            if (tb[t * LT_ + 1] != 0.0f) n++; else break;
        }
        nv[b] = n;
    }
    for (int t = threadIdx.x; t < NT_; t += blockDim.x) {
        const float* tt = tb + t * LT_;
        long bt = (long)b * NT_ + t;
        tclsv[bt] = tt[0];
        float gxk[KC], gyk[KC];
#pragma unroll
        for (int k = 0; k < KC; ++k) {
            gxk[k] = tt[1 + 2 * k];
            gyk[k] = tt[2 + 2 * k];
            gtx[bt * KC + k] = gxk[k];
            gty[bt * KC + k] = gyk[k];
        }
        float gx0 = gxk[0] * (float)NW_;
        float gy0 = gyk[0] * (float)NH_;
        int gi0 = min(max((int)floorf(gx0), 0), NW_ - 1);
        int gj0 = min(max((int)floorf(gy0), 0), NH_ - 1);
        gi0a[bt] = gi0; gj0a[bt] = gj0;
        // anchor IoU argmax (first occurrence of max)
        float gw = tt[LT_ - 2] * (float)NW_;
        float gh = tt[LT_ - 1] * (float)NH_;
        int best = 0; float bestIou = -1.0f;
#pragma unroll
        for (int a = 0; a < NAN_; ++a) {
            float mw = fminf(AW[a], gw), mh = fminf(AH[a], gh);
            float inter = mw * mh;
            float iou = inter / (AW[a] * AH[a] + gw * gh - inter);
            if (iou > bestIou) { bestIou = iou; best = a; }
        }
        bestn[bt] = best;
        // conf_t via wrapped flat index gather (replicates reference's mod trick)
        long off = (long)b * NANCH - NPIX + (long)gj0 * NW_ + gi0;
        long M = (long)NB_ * NANCH;
        long r = ((off % M) + M) % M;
        int bb = (int)(r / NANCH), cc = (int)(r % NANCH);
        float s = 0.0f;
#pragma unroll
        for (int k = 0; k < KC; ++k) {
            float dx = (px[((long)(bb * KC + k)) * NANCH + cc] - gxk[k]) * 640.0f;
            float dy = (py[((long)(bb * KC + k)) * NANCH + cc] - gyk[k]) * 480.0f;
            s += corner_c(dx * dx + dy * dy);
            txv[bt * KC + k] = gxk[k] * (float)NW_ - (float)gi0;
            tyv[bt * KC + k] = gyk[k] * (float)NH_ - (float)gj0;
        }
        conf_t[bt] = s * (1.0f / (float)KC);
    }
}

// ---------------------------------------------------------------------------
// 3) HOT KERNEL: cur_confs max over targets via V_WMMA_F32_16X16X4_F32.
//    d^2(cell,target) = [q,px,py,1] . [1,-2Sx^2 gx,-2Sy^2 gy, qg]  (K=4)
//    One wave: 16 cells x (4 target tiles x 16 targets) x 9 corners.
//    cur_confs is only compared against SIL, so we track the raw corner-sum
//    and compare against SIL*KC (no mean divide in the hot loop).
//    Writes conf_mask0 = (maxconf > SIL) ? 0 : NOOBJ.
// ---------------------------------------------------------------------------
__global__ __launch_bounds__(32)
void conf_mask_kernel(const float* __restrict__ px, const float* __restrict__ py,
                      const float* __restrict__ gtx, const float* __restrict__ gty,
                      const int* __restrict__ nv, float* __restrict__ conf_mask) {
    int b    = blockIdx.x / CELL_TILES;
    int tile = blockIdx.x % CELL_TILES;
    int cellBase = tile * 16;
    int lane = threadIdx.x;
    int half = lane >> 4;        // 0: K=0,1 rows of A / rows 0,1 of B ; 1: K=2,3
    int mrow = lane & 15;

    int cellL = cellBase + mrow;
    int cellC = min(cellL, NANCH - 1);

    // A fragments for all 9 corners (cached in registers)
    float a0k[KC], a1k[KC];
#pragma unroll
    for (int k = 0; k < KC; ++k) {
        float pxx = px[((long)(b * KC + k)) * NANCH + cellC];
        float pyy = py[((long)(b * KC + k)) * NANCH + cellC];
        float q = SX2 * pxx * pxx + SY2 * pyy * pyy;
        a0k[k] = (half == 0) ? q   : pyy;
        a1k[k] = (half == 0) ? pxx : 1.0f;
    }

    int n_valid = nv[b];
    float maxc[8];
#pragma unroll
    for (int r = 0; r < 8; ++r) maxc[r] = 0.0f;

    for (int tt = 0; tt < 4; ++tt) {
        if (tt * 16 >= n_valid) break;
        v8f acc = {0.f, 0.f, 0.f, 0.f, 0.f, 0.f, 0.f, 0.f};
#pragma unroll
        for (int k = 0; k < KC; ++k) {
            int t = tt * 16 + mrow;
            int tc = min(t, NT_ - 1);
            float gx = gtx[((long)b * NT_ + tc) * KC + k];
            float gy = gty[((long)b * NT_ + tc) * KC + k];
            v2f af = {a0k[k], a1k[k]};
            v2f bf;
            if (half == 0) { bf.x = 1.0f;             bf.y = -2.0f * SX2 * gx; }
            else           { bf.x = -2.0f * SY2 * gy; bf.y = SX2 * gx * gx + SY2 * gy * gy; }
            v8f cz = {0.f, 0.f, 0.f, 0.f, 0.f, 0.f, 0.f, 0.f};
            v8f d2 = __builtin_amdgcn_wmma_f32_16x16x4_f32(
                /*neg_a=*/false, af, /*neg_b=*/false, bf,
                /*c_mod=*/(short)0, cz, /*reuse_a=*/false, /*reuse_b=*/false);
#pragma unroll
            for (int r = 0; r < 8; ++r) acc[r] += corner_c(d2[r]);
        }
        // this lane's target column
        int tglob = tt * 16 + mrow;
        bool tok = (tglob < n_valid);
#pragma unroll
        for (int r = 0; r < 8; ++r) {
            float c = tok ? acc[r] : 0.0f;   // un-normalized corner sum
            // max over the 16 target columns (lanes within the same half)
            c = fmaxf(c, __shfl_xor(c, 1));
            c = fmaxf(c, __shfl_xor(c, 2));
            c = fmaxf(c, __shfl_xor(c, 4));
            c = fmaxf(c, __shfl_xor(c, 8));
            maxc[r] = fmaxf(maxc[r], c);
        }
    }

    // C/D layout: element (M = r + 8*half, N = lane&15). After reduction all
    // lanes in a half agree; lane with mrow==0 writes its half's 8 cells.
    if (mrow == 0) {
#pragma unroll
        for (int r = 0; r < 8; ++r) {
            int cell = cellBase + r + 8 * half;
            if (cell < NANCH) {
                // maxc holds sum over K corners; compare vs SIL*KC == mean vs SIL
                conf_mask[(long)b * NANCH + cell] =
                    (maxc[r] > SIL_ * (float)KC) ? 0.0f : NOOBJ_;
            }
        }
    }
}

// ---------------------------------------------------------------------------
// 4) sequential per-batch scatter (replicates scan overwrite semantics)
// ---------------------------------------------------------------------------
__global__ void scatter_kernel(const int* __restrict__ nv,
                               const int* __restrict__ gi0a, const int* __restrict__ gj0a,
                               const int* __restrict__ bestn,
                               const float* __restrict__ conf_t,
                               const float* __restrict__ txv, const float* __restrict__ tyv,
                               const float* __restrict__ tclsv,
                               float* coord_mask, float* cls_mask, float* conf_mask,
                               float* tx, float* ty, float* tconf, float* tcls) {
    int b = blockIdx.x * blockDim.x + threadIdx.x;
    if (b >= NB_) return;
    int n = nv[b];
    for (int t = 0; t < n; ++t) {
        long bt = (long)b * NT_ + t;
        int cell = bestn[bt] * NPIX + gj0a[bt] * NW_ + gi0a[bt];
        long base = (long)b * NANCH + cell;
        coord_mask[base] = 1.0f;
        cls_mask[base]   = 1.0f;
        conf_mask[base]  = OBJ_;
        tconf[base]      = conf_t[bt];
        tcls[base]       = tclsv[bt];
#pragma unroll
        for (int k = 0; k < KC; ++k) {
            tx[base * KC + k] = txv[bt * KC + k];
            ty[base * KC + k] = tyv[bt * KC + k];
        }
    }
}

// ---------------------------------------------------------------------------
// 5) fused loss reduction
// ---------------------------------------------------------------------------
__global__ __launch_bounds__(256)
void loss_kernel(const float* __restrict__ output, const int* __restrict__ epoch_p,
                 const float* __restrict__ coord_mask, const float* __restrict__ cls_mask,
                 const float* __restrict__ conf_mask,
                 const float* __restrict__ tx, const float* __restrict__ ty,
                 const float* __restrict__ tconf, const float* __restrict__ tcls,
                 float* out) {
    long idx = (long)blockIdx.x * blockDim.x + threadIdx.x;
    float lsum = 0.0f;
    if (idx < (long)NCELLS) {
        int b = (int)(idx / NANCH);
        int cell = (int)(idx % NANCH);
        int a = cell / NPIX;
        int pix = cell % NPIX;
        const float* ob = output + ((long)(b * NAN_ + a) * CHAN) * NPIX + pix;
        float cm = coord_mask[idx];
#pragma unroll
        for (int k = 0; k < KC; ++k) {
            float xs = (k == 0) ? sigmoidf_(ob[0])    : ob[(2 * k) * NPIX];
            float ys = (k == 0) ? sigmoidf_(ob[NPIX]) : ob[(2 * k + 1) * NPIX];
            float dx = (xs - tx[idx * KC + k]) * cm;
            float dy = (ys - ty[idx * KC + k]) * cm;
            lsum += 0.5f * (dx * dx + dy * dy);
        }
        // class CE (log_softmax over 13 classes)
        float clm = cls_mask[idx];
        if (clm != 0.0f) {
            float lg[NCLS], m = -3.4e38f;
#pragma unroll
            for (int c = 0; c < NCLS; ++c) { lg[c] = ob[(2 * KC + 1 + c) * NPIX]; m = fmaxf(m, lg[c]); }
            float se = 0.0f;
#pragma unroll
            for (int c = 0; c < NCLS; ++c) se += __builtin_amdgcn_exp2f((lg[c] - m) * LOG2E_);
            float lse = m + LN2_ * __builtin_amdgcn_logf(se);
            int lab = min(max((int)tcls[idx], 0), NCLS - 1);
            lsum += (lse - lg[lab]) * clm;
        }
        if (epoch_p[0] > PRETRAIN) {
            float conf = sigmoidf_(ob[(2 * KC) * NPIX]);
            float dc = conf - tconf[idx];
            lsum += 0.5f * dc * dc * conf_mask[idx];
        }
    }
    __shared__ float red[256];
    red[threadIdx.x] = lsum;
    __syncthreads();
    for (int s = 128; s > 0; s >>= 1) {
        if (threadIdx.x < s) red[threadIdx.x] += red[threadIdx.x + s];
        __syncthreads();
    }
    if (threadIdx.x == 0) atomicAdd(out, red[0]);
}

// ---------------------------------------------------------------------------
extern "C" void kernel_launch(void* const* d_in, const int* in_sizes, int n_in,
                              void* d_out, int out_size, void* d_ws, size_t ws_size,
                              hipStream_t stream) {
    const float* output = (const float*)d_in[0];
    const float* target = (const float*)d_in[1];
    const int*   epoch  = (const int*)d_in[2];
    float* out = (float*)d_out;

    // workspace layout (floats / ints, all 4-byte)
    char* ws = (char*)d_ws;
    size_t off = 0;
    auto allocF = [&](size_t n) { float* p = (float*)(ws + off); off += n * sizeof(float); return p; };
    auto allocI = [&](size_t n) { int*   p = (int*)(ws + off);   off += n * sizeof(int);   return p; };

    float* px    = allocF((size_t)NB_ * KC * NANCH);
    float* py    = allocF((size_t)NB_ * KC * NANCH);
    float* gtx   = allocF((size_t)NB_ * NT_ * KC);
    float* gty   = allocF((size_t)NB_ * NT_ * KC);
    float* conf_t= allocF((size_t)NB_ * NT_);
    float* txv   = allocF((size_t)NB_ * NT_ * KC);
    float* tyv   = allocF((size_t)NB_ * NT_ * KC);
    float* tclsv = allocF((size_t)NB_ * NT_);
    float* conf_mask  = allocF((size_t)NCELLS);
    float* coord_mask = allocF((size_t)NCELLS);
    float* cls_mask   = allocF((size_t)NCELLS);
    float* tconf      = allocF((size_t)NCELLS);
    float* tcls       = allocF((size_t)NCELLS);
    float* tx         = allocF((size_t)NCELLS * KC);
    float* ty         = allocF((size_t)NCELLS * KC);
    int* nv    = allocI(NB_);
    int* gi0a  = allocI((size_t)NB_ * NT_);
    int* gj0a  = allocI((size_t)NB_ * NT_);
    int* bestn = allocI((size_t)NB_ * NT_);
    (void)ws_size; (void)in_sizes; (void)n_in; (void)out_size;

    {   // zero-init sparse-scatter targets + output scalar
        long n = (long)NCELLS * KC;
        int blocks = (int)((n + 255) / 256);
        init_kernel<<<blocks, 256, 0, stream>>>(coord_mask, cls_mask, tconf, tcls, tx, ty, out);
    }
    {
        int blocks = (NCELLS + 255) / 256;
        compute_pred_kernel<<<blocks, 256, 0, stream>>>(output, px, py);
    }
    prep_targets_kernel<<<NB_, 64, 0, stream>>>(target, px, py, gtx, gty, nv,
                                                gi0a, gj0a, bestn, conf_t, txv, tyv, tclsv);
    conf_mask_kernel<<<NB_ * CELL_TILES, 32, 0, stream>>>(px, py, gtx, gty, nv, conf_mask);
    scatter_kernel<<<1, NB_, 0, stream>>>(nv, gi0a, gj0a, bestn, conf_t, txv, tyv, tclsv,
                                          coord_mask, cls_mask, conf_mask, tx, ty, tconf, tcls);
    {
        int blocks = (NCELLS + 255) / 256;
        loss_kernel<<<blocks, 256, 0, stream>>>(output, epoch, coord_mask, cls_mask, conf_mask,
                                                tx, ty, tconf, tcls, out);
    }
}